// AttentionLayer_70282844831888
// MI455X (gfx1250) — compile-verified
//
#include <hip/hip_runtime.h>
#include <hip/hip_bf16.h>

typedef __attribute__((ext_vector_type(2))) float v2f;
typedef __attribute__((ext_vector_type(8))) float v8f;

#define BB 2
#define SS 512
#define DD 512
#define HH 256

// gfx1250 hardware tanh (TRANS unit, co-executes with VALU) if available.
__device__ __forceinline__ float hw_tanh(float x) {
#if __has_builtin(__builtin_amdgcn_tanhf)
  return __builtin_amdgcn_tanhf(x);
#else
  return tanhf(x);
#endif
}

// One wave computes a 16x64 strip of C = A[M,K] @ B[K,N] (+bias), all f32,
// via V_WMMA_F32_16X16X4_F32: 4 accumulators share each A fetch (4x A reuse).
// A (16x4 f32): lanes<16 hold K=kg*2,kg*2+1 for row M=lane%16 (kg = lane/16).
// B (4x16 f32): VGPR0/1 = rows K=kg*2, kg*2+1; lanes%16 = N.
// C/D: VGPR r -> M = r + kg*8, N = lane%16.
__device__ __forceinline__ void wmma_f32_strip(const float* __restrict__ A, int lda,
                                               const float* __restrict__ Bm, int ldb,
                                               float* __restrict__ C, int ldc,
                                               const float* __restrict__ bias,
                                               int m0, int n0, int K, int lane) {
  v8f acc0 = {}, acc1 = {}, acc2 = {}, acc3 = {};
  const int mrow = lane & 15;
  const int kg   = (lane >> 4) & 1;
  const float* Ap = A + (size_t)(m0 + mrow) * lda + kg * 2;
  const float* Bp = Bm + (size_t)(kg * 2) * ldb + n0 + mrow;
  for (int k0 = 0; k0 < K; k0 += 4) {
    v2f av, b0, b1, b2, b3;
    av.x = Ap[k0];
    av.y = Ap[k0 + 1];
    const float* Br0 = Bp + (size_t)k0 * ldb;
    const float* Br1 = Br0 + ldb;
    b0.x = Br0[0];  b0.y = Br1[0];
    b1.x = Br0[16]; b1.y = Br1[16];
    b2.x = Br0[32]; b2.y = Br1[32];
    b3.x = Br0[48]; b3.y = Br1[48];
    acc0 = __builtin_amdgcn_wmma_f32_16x16x4_f32(false, av, false, b0, (short)0, acc0, false, false);
    acc1 = __builtin_amdgcn_wmma_f32_16x16x4_f32(false, av, false, b1, (short)0, acc1, false, false);
    acc2 = __builtin_amdgcn_wmma_f32_16x16x4_f32(false, av, false, b2, (short)0, acc2, false, false);
    acc3 = __builtin_amdgcn_wmma_f32_16x16x4_f32(false, av, false, b3, (short)0, acc3, false, false);
  }
  v8f accs[4] = {acc0, acc1, acc2, acc3};
#pragma unroll
  for (int u = 0; u < 4; ++u) {
    const int nc = n0 + u * 16 + mrow;
    const float badd = bias ? bias[nc] : 0.0f;
#pragma unroll
    for (int r = 0; r < 8; ++r) {
      C[(size_t)(m0 + r + kg * 8) * ldc + nc] = accs[u][r] + badd;
    }
  }
}

// Kernel 1: a = x@w1 + b1 and t = x@w2 + b2 (grid.y selects which).
// x viewed as [B*S, D] row-major; W is [D, H]; out is [B*S, H].
__global__ void k_gemm_bias(const float* __restrict__ x,
                            const float* __restrict__ w1, const float* __restrict__ b1,
                            const float* __restrict__ w2, const float* __restrict__ b2,
                            float* __restrict__ a_buf, float* __restrict__ t_buf) {
  const int lane = threadIdx.x & 31;
  const int wave = threadIdx.x >> 5;                 // 4 waves / block
  const int tile = blockIdx.x * 4 + wave;            // wave-uniform
  const int tiles_n = HH / 64;                       // 4 strips of width 64
  const int tm = tile / tiles_n;
  const int tn = tile % tiles_n;
  if (tm * 16 >= BB * SS) return;                    // wave-uniform branch
  const float* W    = blockIdx.y ? w2 : w1;
  const float* bias = blockIdx.y ? b2 : b1;
  float* out        = blockIdx.y ? t_buf : a_buf;
  wmma_f32_strip(x, DD, W, HH, out, HH, bias, tm * 16, tn * 64, DD, lane);
}

// Kernel 2: p[b,i,j] from scores[b,i,j] = sum_h tanh(a[b,j,h]+t[b,i,h])*v[h] + bv.
// 16x16 (i,j) tile per block, 256 threads, one (i,j) per thread.
// LDS stride 257 (257 % 64 == 1) => bank = (row + h) % 64, conflict-free.
__global__ void k_scores(const float* __restrict__ a_buf,
                         const float* __restrict__ t_buf,
                         const float* __restrict__ v,
                         const float* __restrict__ bv_p,
                         const int* __restrict__ mask,
                         float* __restrict__ p_buf) {
  __shared__ float aS[16 * 257];
  __shared__ float tS[16 * 257];
  __shared__ float vS[HH];
  const int b  = blockIdx.z;
  const int i0 = blockIdx.y * 16;
  const int j0 = blockIdx.x * 16;
  const int tid = threadIdx.x;    // 256

  for (int idx = tid; idx < 16 * HH; idx += 256) {
    const int r = idx >> 8;       // HH == 256
    const int c = idx & (HH - 1);
    aS[r * 257 + c] = a_buf[((size_t)b * SS + j0 + r) * HH + c];
    tS[r * 257 + c] = t_buf[((size_t)b * SS + i0 + r) * HH + c];
  }
  vS[tid & (HH - 1)] = v[tid & (HH - 1)];
  __syncthreads();

  const int i = tid >> 4;
  const int j = tid & 15;
  const float* __restrict__ ar = &aS[j * 257];
  const float* __restrict__ tr = &tS[i * 257];
  float s = 0.0f;
#pragma unroll 8
  for (int h = 0; h < HH; ++h) {
    s += hw_tanh(ar[h] + tr[h]) * vS[h];
  }
  s += bv_p[0];
  const float m = (float)mask[(size_t)b * SS + j0 + j];
  const float e = expf(s) * m;
  const float pv = e / (e + 1e-16f);
  p_buf[((size_t)b * SS + i0 + i) * SS + j0 + j] = pv;
}

// Kernel 3: out[b] = p[b] @ x[b]   ([S,S]@[S,D] f32 WMMA).
__global__ void k_gemm_out(const float* __restrict__ p_buf,
                           const float* __restrict__ x,
                           float* __restrict__ out) {
  const int lane = threadIdx.x & 31;
  const int wave = threadIdx.x >> 5;
  const int b    = blockIdx.y;
  const int tile = blockIdx.x * 4 + wave;
  const int tiles_n = DD / 64;                       // 8 strips of width 64
  const int tm = tile / tiles_n;
  const int tn = tile % tiles_n;
  if (tm * 16 >= SS) return;                         // wave-uniform
  wmma_f32_strip(p_buf + (size_t)b * SS * SS, SS,
                 x     + (size_t)b * SS * DD, DD,
                 out   + (size_t)b * SS * DD, DD,
                 nullptr, tm * 16, tn * 64, SS, lane);
}

extern "C" void kernel_launch(void* const* d_in, const int* in_sizes, int n_in,
                              void* d_out, int out_size, void* d_ws, size_t ws_size,
                              hipStream_t stream) {
  const float* x    = (const float*)d_in[0];
  const float* w1   = (const float*)d_in[1];
  const float* b1   = (const float*)d_in[2];
  const float* w2   = (const float*)d_in[3];
  const float* b2   = (const float*)d_in[4];
  const float* v    = (const float*)d_in[5];
  const float* bv   = (const float*)d_in[6];
  const int*   mask = (const int*)d_in[7];
  float* out = (float*)d_out;

  float* a_buf = (float*)d_ws;                       // [B*S, H]  1 MB
  float* t_buf = a_buf + (size_t)BB * SS * HH;       // [B*S, H]  1 MB
  float* p_buf = t_buf + (size_t)BB * SS * HH;       // [B,S,S]   2 MB

  // Kernel 1: 256 strips (64 x 4), 4 waves/block -> 64 blocks; y = {w1,w2}.
  {
    dim3 grid(64, 2, 1);
    dim3 block(128, 1, 1);
    k_gemm_bias<<<grid, block, 0, stream>>>(x, w1, b1, w2, b2, a_buf, t_buf);
  }
  // Kernel 2: (S/16) x (S/16) x B blocks, 256 threads.
  {
    dim3 grid(SS / 16, SS / 16, BB);
    dim3 block(256, 1, 1);
    k_scores<<<grid, block, 0, stream>>>(a_buf, t_buf, v, bv, mask, p_buf);
  }
  // Kernel 3: 256 strips (32 x 8) per batch, 4 waves/block -> 64 blocks; y = batch.
  {
    dim3 grid(64, BB, 1);
    dim3 block(128, 1, 1);
    k_gemm_out<<<grid, block, 0, stream>>>(p_buf, x, out);
  }
}